// MesaLayerMemory_62586263437944
// MI455X (gfx1250) — compile-verified
//
#include <hip/hip_runtime.h>

typedef __attribute__((ext_vector_type(2))) float v2f;
typedef __attribute__((ext_vector_type(8))) float v8f;
typedef __attribute__((ext_vector_type(4))) unsigned int u32x4;
typedef __attribute__((ext_vector_type(4))) int i32x4;
typedef __attribute__((ext_vector_type(8))) int i32x8;

#define B_   64
#define T_   2048
#define DK_  128
#define DV_  128
#define NQ_  2048
#define LAM_ 1.0f
#define EPS_ 1e-6f

// ---------------------------------------------------------------------------
// Tensor Data Mover: issue one 2D tile load (global -> LDS).
// D# built per CDNA5 ISA ch.8: group0 = {ctrl, lds_addr, gaddr_lo, gaddr_hi|type},
// group1 = {mask/dsize/pad, tensor dims, tile dims, dim0 stride}; groups 2/3
// zero (2D tensor). data_size = 4 bytes. All dims in elements.
// pad_word: optional LDS-padding control bits ([20] enable, [24:22] interval,
// [31:25] amount) to produce bank-padded LDS rows straight out of the DMA.
// ---------------------------------------------------------------------------
__device__ __forceinline__ void tdm_load_2d(unsigned lds_addr, const void* gptr,
                                            unsigned tensor_d0, unsigned tensor_d1,
                                            unsigned tile_d0, unsigned tile_d1,
                                            unsigned stride_d0, unsigned pad_word) {
  const unsigned long long ga = (unsigned long long)(uintptr_t)gptr;
  u32x4 g0;
  g0[0] = 1u;                                   // count=1 (valid user descriptor)
  g0[1] = lds_addr;                             // LDS byte address
  g0[2] = (unsigned)ga;                         // global_addr[31:0]
  g0[3] = (unsigned)(ga >> 32) | (2u << 30);    // global_addr[56:32] | type=2
  i32x8 g1;
  g1[0] = (int)(0x00020000u | pad_word);        // wg_mask=0, data_size=4B, pad ctrl
  g1[1] = (int)(tensor_d0 << 16);               // tensor_dim0[15:0]
  g1[2] = (int)((tensor_d0 >> 16) | (tensor_d1 << 16));
  g1[3] = (int)((tensor_d1 >> 16) | (tile_d0 << 16));
  g1[4] = (int)(tile_d1 & 0xFFFFu);             // tile_dim1 (tile_dim2 = 0)
  g1[5] = (int)stride_d0;                       // tensor_dim0_stride[31:0]
  g1[6] = 0;
  g1[7] = 0;
  i32x4 z4 = {0, 0, 0, 0};
#if defined(__clang_major__) && (__clang_major__ >= 23)
  i32x8 z8 = {0, 0, 0, 0, 0, 0, 0, 0};
  __builtin_amdgcn_tensor_load_to_lds(g0, g1, z4, z4, z8, 0);
#else
  __builtin_amdgcn_tensor_load_to_lds(g0, g1, z4, z4, 0);
#endif
}

// ---------------------------------------------------------------------------
// Kernel 1: suffix products of gammas.  w[b][t] = prod_{s>t} g[b][s]
// ---------------------------------------------------------------------------
__global__ void __launch_bounds__(256)
gamma_suffix_kernel(const float* __restrict__ g, float* __restrict__ w) {
  const int b   = blockIdx.x;
  const int tid = threadIdx.x;
  const float* gb = g + (size_t)b * T_;
  float*       wb = w + (size_t)b * T_;

  float loc[8];
  float p = 1.0f;
#pragma unroll
  for (int i = 0; i < 8; ++i) loc[i] = gb[tid * 8 + i];
#pragma unroll
  for (int i = 0; i < 8; ++i) p *= loc[i];

  __shared__ float chunk[256];
  chunk[tid] = p;
  __syncthreads();
  for (int off = 1; off < 256; off <<= 1) {
    float v = (tid + off < 256) ? chunk[tid + off] : 1.0f;
    __syncthreads();
    chunk[tid] *= v;
    __syncthreads();
  }
  float suf = (tid == 255) ? 1.0f : chunk[tid + 1];
#pragma unroll
  for (int i = 7; i >= 0; --i) {
    wb[tid * 8 + i] = suf;
    suf *= loc[i];
  }
}

// ---------------------------------------------------------------------------
// Kernel 2: exact Sherman-Morrison scan for R (LDS-resident, one block/batch)
// ---------------------------------------------------------------------------
__global__ void __launch_bounds__(256)
r_scan_kernel(const float* __restrict__ keys, float* __restrict__ Rout) {
  const int b   = blockIdx.x;
  const int tid = threadIdx.x;
  const float* Kb = keys + (size_t)b * T_ * DK_;

  __shared__ float R[DK_][DK_ + 1];
  __shared__ float kv[DK_];
  __shared__ float Rk[DK_];
  __shared__ float partial[256];
  __shared__ float inv_denom;

  for (int idx = tid; idx < DK_ * DK_; idx += 256) {
    int i = idx >> 7, j = idx & (DK_ - 1);
    R[i][j] = (i == j) ? LAM_ : 0.0f;
  }
  __syncthreads();

  const int row = tid >> 1;
  const int h   = tid & 1;

  for (int t = 0; t < T_; ++t) {
    if (tid < DK_) kv[tid] = Kb[(size_t)t * DK_ + tid];
    __syncthreads();

    {
      const float* Rr = &R[row][h * 64];
      const float* kk = &kv[h * 64];
      float s = 0.0f;
#pragma unroll 16
      for (int j = 0; j < 64; ++j) s += Rr[j] * kk[j];
      partial[tid] = s;
    }
    __syncthreads();
    if (tid < DK_) Rk[tid] = partial[2 * tid] + partial[2 * tid + 1];
    __syncthreads();

    if (tid < 32) {
      float s = 0.0f;
#pragma unroll
      for (int j = 0; j < 4; ++j) s += kv[tid * 4 + j] * Rk[tid * 4 + j];
      partial[tid] = s;
    }
    __syncthreads();
    if (tid == 0) {
      float s = 0.0f;
#pragma unroll
      for (int j = 0; j < 32; ++j) s += partial[j];
      inv_denom = 1.0f / (1.0f + s + EPS_);
    }
    __syncthreads();

    {
      const float ri = Rk[row] * inv_denom;
      float*       Rr = &R[row][h * 64];
      const float* rj = &Rk[h * 64];
#pragma unroll 16
      for (int j = 0; j < 64; ++j) Rr[j] -= ri * rj[j];
    }
    __syncthreads();
  }

  float* Ro = Rout + (size_t)b * DK_ * DK_;
  for (int idx = tid; idx < DK_ * DK_; idx += 256)
    Ro[idx] = R[idx >> 7][idx & (DK_ - 1)];
}

// ---------------------------------------------------------------------------
// Kernel 3 (fused): S = (w .* V)^T K with TDM-staged K tiles (double-buffered
// LDS), then phi = S @ R.  One block per batch, 8 waves, WMMA f32 16x16x4.
// ---------------------------------------------------------------------------
__global__ void __launch_bounds__(256)
s_phi_kernel(const float* __restrict__ keys, const float* __restrict__ values,
             const float* __restrict__ w, const float* __restrict__ R,
             float* __restrict__ phi) {
  const int b     = blockIdx.x;
  const int wave  = threadIdx.x >> 5;
  const int lane  = threadIdx.x & 31;
  const int lhalf = lane >> 4;
  const int l16   = lane & 15;
  const int m0    = wave * 16;

  __shared__ float Ssh[DV_][130];      // padded rows for phase-2 column reads
  __shared__ float Kt[2][32][DK_];     // double-buffered K chunks (TDM dest)

  const float* Kb = keys   + (size_t)b * T_ * DK_;
  const float* Vb = values + (size_t)b * T_ * DV_;
  const float* wb = w      + (size_t)b * T_;

  v8f acc[8];
#pragma unroll
  for (int n = 0; n < 8; ++n)
#pragma unroll
    for (int r = 0; r < 8; ++r) acc[n][r] = 0.0f;

  const int NCH = T_ / 32;
  if (wave == 0) {
    tdm_load_2d((unsigned)(uintptr_t)&Kt[0][0][0], Kb,
                DK_, T_, DK_, 32, DK_, 0);
    tdm_load_2d((unsigned)(uintptr_t)&Kt[1][0][0], Kb + 32 * DK_,
                DK_, T_, DK_, 32, DK_, 0);
  }

  // ---- Phase 1: S[m0:m0+16][:] = sum_t w_t * V[t][m] * K[t][n] ----
  for (int c = 0; c < NCH; ++c) {
    if (wave == 0) __builtin_amdgcn_s_wait_tensorcnt(1);  // oldest chunk landed
    __syncthreads();
    const float (*Kc)[DK_] = Kt[c & 1];
    const int tb = c * 32;
#pragma unroll 2
    for (int tl = 0; tl < 32; tl += 4) {
      const int ta  = tb + tl + 2 * lhalf;
      const int tla = tl + 2 * lhalf;
      v2f a;
      a[0] = wb[ta]     * Vb[(size_t)ta * DV_ + m0 + l16];
      a[1] = wb[ta + 1] * Vb[(size_t)(ta + 1) * DV_ + m0 + l16];
#pragma unroll
      for (int n = 0; n < 8; ++n) {
        v2f bb;
        bb[0] = Kc[tla][n * 16 + l16];
        bb[1] = Kc[tla + 1][n * 16 + l16];
        acc[n] = __builtin_amdgcn_wmma_f32_16x16x4_f32(
            false, a, false, bb, (short)0, acc[n], false, false);
      }
    }
    __syncthreads();
    if (wave == 0 && c + 2 < NCH)
      tdm_load_2d((unsigned)(uintptr_t)&Kt[c & 1][0][0],
                  Kb + (size_t)(c + 2) * 32 * DK_,
                  DK_, T_, DK_, 32, DK_, 0);
  }

  // spill S tiles to LDS
#pragma unroll
  for (int n = 0; n < 8; ++n)
#pragma unroll
    for (int r = 0; r < 8; ++r)
      Ssh[m0 + lhalf * 8 + r][n * 16 + l16] = acc[n][r];
  __syncthreads();

  // ---- Phase 2: phi[m0:m0+16][:] = S[m0:m0+16][:] @ R ----
  const float* Rb = R + (size_t)b * DK_ * DK_;
#pragma unroll
  for (int n = 0; n < 8; ++n)
#pragma unroll
    for (int r = 0; r < 8; ++r) acc[n][r] = 0.0f;

  for (int j = 0; j < DK_; j += 4) {
    const int ja = j + 2 * lhalf;
    v2f a = *(const v2f*)&Ssh[m0 + l16][ja];   // ds_load_b64
#pragma unroll
    for (int n = 0; n < 8; ++n) {
      const int n0 = n * 16;
      v2f bb;
      bb[0] = Rb[(size_t)ja * DK_ + n0 + l16];
      bb[1] = Rb[(size_t)(ja + 1) * DK_ + n0 + l16];
      acc[n] = __builtin_amdgcn_wmma_f32_16x16x4_f32(
          false, a, false, bb, (short)0, acc[n], false, false);
    }
  }

  float* pb = phi + (size_t)b * DV_ * DK_;
#pragma unroll
  for (int n = 0; n < 8; ++n)
#pragma unroll
    for (int r = 0; r < 8; ++r)
      pb[(size_t)(m0 + lhalf * 8 + r) * DK_ + n * 16 + l16] = acc[n][r];
}

// ---------------------------------------------------------------------------
// Kernel 4: out[b] = Q[b] @ phi[b]^T.  phi staged by a single TDM tile load
// with LDS padding (2 DWORDs every 128 -> [128][130] bank-friendly layout).
// ---------------------------------------------------------------------------
__global__ void __launch_bounds__(256)
readout_kernel(const float* __restrict__ Q, const float* __restrict__ phi,
               float* __restrict__ out) {
  const int b    = blockIdx.y;
  const int mblk = blockIdx.x;

  __shared__ float P[DV_][130];
  if ((threadIdx.x >> 5) == 0) {
    // pad_enable | interval=code6 (128 DWORDs) | amount=code1 (2 DWORDs)
    const unsigned pad = (1u << 20) | (6u << 22) | (1u << 25);
    tdm_load_2d((unsigned)(uintptr_t)&P[0][0], phi + (size_t)b * DV_ * DK_,
                DK_, DV_, DK_, DV_, DK_, pad);
  }
  __builtin_amdgcn_s_wait_tensorcnt(0);
  __syncthreads();

  const int wave  = threadIdx.x >> 5;
  const int lane  = threadIdx.x & 31;
  const int lhalf = lane >> 4;
  const int l16   = lane & 15;
  const int m0    = mblk * 128 + wave * 16;

  const float* qb = Q + (size_t)b * NQ_ * DK_;

  v8f acc[8];
#pragma unroll
  for (int n = 0; n < 8; ++n)
#pragma unroll
    for (int r = 0; r < 8; ++r) acc[n][r] = 0.0f;

  for (int j = 0; j < DK_; j += 4) {
    const int ja = j + 2 * lhalf;
    v2f a = *(const v2f*)(qb + (size_t)(m0 + l16) * DK_ + ja);  // global_load_b64
#pragma unroll
    for (int n = 0; n < 8; ++n) {
      const int n0 = n * 16;
      v2f bb;                                // B[k][v] = phi[v][k]
      bb[0] = P[n0 + l16][ja];
      bb[1] = P[n0 + l16][ja + 1];
      acc[n] = __builtin_amdgcn_wmma_f32_16x16x4_f32(
          false, a, false, bb, (short)0, acc[n], false, false);
    }
  }

  float* ob = out + (size_t)b * NQ_ * DV_;
#pragma unroll
  for (int n = 0; n < 8; ++n)
#pragma unroll
    for (int r = 0; r < 8; ++r)
      ob[(size_t)(m0 + lhalf * 8 + r) * DV_ + n * 16 + l16] = acc[n][r];
}

// ---------------------------------------------------------------------------
extern "C" void kernel_launch(void* const* d_in, const int* in_sizes, int n_in,
                              void* d_out, int out_size, void* d_ws, size_t ws_size,
                              hipStream_t stream) {
  const float* keys    = (const float*)d_in[0];
  const float* values  = (const float*)d_in[1];
  const float* gammas  = (const float*)d_in[2];
  const float* queries = (const float*)d_in[3];
  float* out = (float*)d_out;

  float* ws  = (float*)d_ws;
  float* w   = ws;                                   // B*T
  float* R   = w + (size_t)B_ * T_;                  // B*128*128
  float* phi = R + (size_t)B_ * DK_ * DK_;           // B*128*128

  gamma_suffix_kernel<<<B_, 256, 0, stream>>>(gammas, w);
  r_scan_kernel<<<B_, 256, 0, stream>>>(keys, R);
  s_phi_kernel<<<B_, 256, 0, stream>>>(keys, values, w, R, phi);
  readout_kernel<<<dim3(NQ_ / 128, B_), 256, 0, stream>>>(queries, phi, out);
}